// PointNetSetAbstraction_523986010540
// MI455X (gfx1250) — compile-verified
//
#include <hip/hip_runtime.h>
#include <cmath>

typedef float v2f __attribute__((ext_vector_type(2)));
typedef float v8f __attribute__((ext_vector_type(8)));

#define BATCH   8
#define NPTS    4096
#define SPTS    1024
#define KSAMP   32
#define CIN     64
#define TDIM    128
#define MROWS   (BATCH * SPTS * KSAMP)   // 262144
#define RAD2    0.04f
#define BN_EPS  1e-5f

__device__ __forceinline__ float gelu_f(float x) {
    return 0.5f * x * (1.0f + erff(x * 0.70710678118654752f));
}

// ---------------------------------------------------------------------------
// 1) Farthest point sampling: one block per batch. xyz staged in LDS,
//    distances in registers, wave32 shuffle + LDS argmax reduction.
// ---------------------------------------------------------------------------
__global__ __launch_bounds__(256) void fps_kernel(const float* __restrict__ xyz,
                                                  float* __restrict__ new_xyz) {
    __shared__ float lx[NPTS], ly[NPTS], lz[NPTS];
    __shared__ int   fidx[SPTS];
    __shared__ float rbd[8];
    __shared__ int   rbi[8];
    __shared__ int   cur_s;

    const int b = blockIdx.x, tid = threadIdx.x;
    float dist[16], px[16], py[16], pz[16];

    #pragma unroll
    for (int j = 0; j < 16; ++j) {
        int i = j * 256 + tid;
        size_t base = ((size_t)b * NPTS + i) * 3;
        float x = xyz[base + 0], y = xyz[base + 1], z = xyz[base + 2];
        px[j] = x; py[j] = y; pz[j] = z;
        lx[i] = x; ly[i] = y; lz[i] = z;
        dist[j] = 1e10f;
    }
    if (tid == 0) { fidx[0] = 0; cur_s = 0; }
    __syncthreads();

    for (int it = 0; it < SPTS; ++it) {
        int cur = cur_s;
        float cx = lx[cur], cy = ly[cur], cz = lz[cur];
        float bd = -1.0f; int bi = 0x7fffffff;
        #pragma unroll
        for (int j = 0; j < 16; ++j) {
            float dx = px[j] - cx, dy = py[j] - cy, dz = pz[j] - cz;
            float d = dx * dx + dy * dy + dz * dz;
            float nd = fminf(dist[j], d);
            dist[j] = nd;
            int i = j * 256 + tid;
            if (nd > bd || (nd == bd && i < bi)) { bd = nd; bi = i; }
        }
        // wave32 reduce (argmax, tie -> smaller index)
        #pragma unroll
        for (int off = 16; off > 0; off >>= 1) {
            float od = __shfl_down(bd, off);
            int   oi = __shfl_down(bi, off);
            if (od > bd || (od == bd && oi < bi)) { bd = od; bi = oi; }
        }
        if ((tid & 31) == 0) { rbd[tid >> 5] = bd; rbi[tid >> 5] = bi; }
        __syncthreads();
        if (tid == 0) {
            float xbd = rbd[0]; int xbi = rbi[0];
            #pragma unroll
            for (int q = 1; q < 8; ++q)
                if (rbd[q] > xbd || (rbd[q] == xbd && rbi[q] < xbi)) { xbd = rbd[q]; xbi = rbi[q]; }
            if (it < SPTS - 1) fidx[it + 1] = xbi;
            cur_s = xbi;
        }
        __syncthreads();
    }
    for (int s = tid; s < SPTS; s += 256) {
        int id = fidx[s];
        size_t o = ((size_t)b * SPTS + s) * 3;
        new_xyz[o + 0] = lx[id];
        new_xyz[o + 1] = ly[id];
        new_xyz[o + 2] = lz[id];
    }
}

// ---------------------------------------------------------------------------
// 2) tp[b,c] = sum_t gelu(te[b,t]) * wt[c,t] + bt[c]
// ---------------------------------------------------------------------------
__global__ void tp_kernel(const float* __restrict__ te, const float* __restrict__ wt,
                          const float* __restrict__ bt, float* __restrict__ tp, int ci) {
    int id = blockIdx.x * blockDim.x + threadIdx.x;
    if (id >= BATCH * ci) return;
    int b = id / ci, c = id % ci;
    float acc = bt[c];
    for (int t = 0; t < TDIM; ++t)
        acc += gelu_f(te[b * TDIM + t]) * wt[c * TDIM + t];
    tp[id] = acc;
}

// 3) bias_eff[b,o] = bc[o] + sum_c tp[b,c] * wc[o,c]
__global__ void bias_kernel(const float* __restrict__ tp, const float* __restrict__ wc,
                            const float* __restrict__ bc, float* __restrict__ bias,
                            int ci, int co) {
    int id = blockIdx.x * blockDim.x + threadIdx.x;
    if (id >= BATCH * co) return;
    int b = id / co, o = id % co;
    float acc = bc[o];
    for (int c = 0; c < ci; ++c)
        acc += tp[b * ci + c] * wc[o * ci + c];
    bias[id] = acc;
}

// ---------------------------------------------------------------------------
// 4) Ball query (first-32-in-index-order semantics) + gather into A0 (ld=68)
//    One wave32 per centroid.
// ---------------------------------------------------------------------------
__global__ __launch_bounds__(256) void ball_gather_kernel(const float* __restrict__ xyz,
                                                          const float* __restrict__ pts,
                                                          const float* __restrict__ nxyz,
                                                          float* __restrict__ A0) {
    __shared__ int sidx[8][KSAMP];
    const int w = threadIdx.x >> 5, lane = threadIdx.x & 31;
    const int gid = blockIdx.x * 8 + w;          // centroid id in [0, B*S)
    const int b = gid >> 10, s = gid & 1023;

    size_t nb = ((size_t)b * SPTS + s) * 3;
    float nx = nxyz[nb + 0], ny = nxyz[nb + 1], nz = nxyz[nb + 2];

    int cnt = 0;
    for (int base = 0; base < NPTS && cnt < KSAMP; base += 32) {
        int i = base + lane;
        size_t pb = ((size_t)b * NPTS + i) * 3;
        float dx = xyz[pb + 0] - nx, dy = xyz[pb + 1] - ny, dz = xyz[pb + 2] - nz;
        bool in_r = (dx * dx + dy * dy + dz * dz) <= RAD2;
        unsigned mask = (unsigned)__ballot(in_r);
        int pos = cnt + __popc(mask & ((1u << lane) - 1u));
        if (in_r && pos < KSAMP) sidx[w][pos] = i;
        cnt += __popc(mask);
    }
    asm volatile("s_wait_dscnt 0" ::: "memory");
    if (cnt < KSAMP) {
        int first = sidx[w][0];
        if (lane >= cnt) sidx[w][lane] = first;
    }
    asm volatile("s_wait_dscnt 0" ::: "memory");

    int id = sidx[w][lane];
    size_t row = (size_t)gid * KSAMP + lane;
    size_t pb = ((size_t)b * NPTS + id) * 3;
    A0[row * 68 + 0] = xyz[pb + 0] - nx;
    A0[row * 68 + 1] = xyz[pb + 1] - ny;
    A0[row * 68 + 2] = xyz[pb + 2] - nz;
    A0[row * 68 + 67] = 0.0f;                    // K padding column
    for (int c = 0; c < CIN; ++c)
        A0[row * 68 + 3 + c] = pts[((size_t)b * CIN + c) * NPTS + id];
}

// ---------------------------------------------------------------------------
// 5) GEMM (M x KPAD) * Wt (KPAD x NC) + bias_eff[b] via V_WMMA_F32_16X16X4_F32
//    + per-channel sum/sumsq partials for BatchNorm (ds atomics -> global).
//    Block = 128 threads (4 waves), 64 rows per block, all NC columns.
//    Weights staged in LDS *pair-interleaved* so each B fragment is one
//    contiguous 8-byte ds_load_b64: Wsp[((k>>1)*NC + n)*2 + (k&1)] = W^T[k][n].
// ---------------------------------------------------------------------------
template <int KC, int KPAD, int NC>
__global__ __launch_bounds__(128) void gemm_bn_kernel(const float* __restrict__ A,
                                                      const float* __restrict__ W,
                                                      const float* __restrict__ biasEff,
                                                      float* __restrict__ Y,
                                                      float* __restrict__ gsum,
                                                      float* __restrict__ gsq) {
    constexpr int ROWS = 64;
    __shared__ __align__(16) float As[ROWS * KPAD];
    __shared__ __align__(16) float Wsp[KPAD * NC];   // pair-interleaved W^T
    __shared__ float bsum[NC], bsq[NC];

    const int tid = threadIdx.x;
    const int rows0 = blockIdx.x * ROWS;

    for (int t = tid; t < KPAD * NC; t += 128) {
        int k = t / NC, n = t % NC;
        float v = (k < KC) ? W[n * KC + k] : 0.0f;
        Wsp[((k >> 1) * NC + n) * 2 + (k & 1)] = v;
    }
    for (int t = tid; t < ROWS * KPAD; t += 128)
        As[t] = A[(size_t)rows0 * KPAD + t];
    for (int t = tid; t < NC; t += 128) { bsum[t] = 0.0f; bsq[t] = 0.0f; }
    __syncthreads();

    const int w = tid >> 5, lane = tid & 31;
    const int hi = lane >> 4, lo = lane & 15;
    const int mrow = w * 16 + lo;                  // A frag: lanes 0-15/16-31 both M=0..15

    v8f acc[NC / 16];
    #pragma unroll
    for (int t = 0; t < NC / 16; ++t) acc[t] = (v8f)0.0f;

    #pragma unroll
    for (int k = 0; k < KPAD; k += 4) {
        // A 16x4 f32 frag: vgpr0 = K=k+2*hi, vgpr1 = K=k+2*hi+1 (contiguous in LDS)
        v2f a = *(const v2f*)&As[mrow * KPAD + k + 2 * hi];
        const int kp = (k >> 1) + hi;              // pair index for this lane half
        #pragma unroll
        for (int t = 0; t < NC / 16; ++t) {
            int n = t * 16 + lo;
            v2f bf = *(const v2f*)&Wsp[(kp * NC + n) * 2];   // one ds_load_b64
            acc[t] = __builtin_amdgcn_wmma_f32_16x16x4_f32(
                false, a, false, bf, (short)0, acc[t], false, false);
        }
    }

    const int b = rows0 / (SPTS * KSAMP);          // 64 | 32768, never crosses batches
    #pragma unroll
    for (int t = 0; t < NC / 16; ++t) {
        int n = t * 16 + lo;
        float bias = biasEff[b * NC + n];
        float s1 = 0.0f, s2 = 0.0f;
        #pragma unroll
        for (int r = 0; r < 8; ++r) {
            float v = acc[t][r] + bias;            // C layout: col = lo, row = r + 8*hi
            int row = rows0 + w * 16 + r + 8 * hi;
            Y[(size_t)row * NC + n] = v;
            s1 += v; s2 += v * v;
        }
        atomicAdd(&bsum[n], s1);
        atomicAdd(&bsq[n], s2);
    }
    __syncthreads();
    for (int t = tid; t < NC; t += 128) {
        atomicAdd(&gsum[t], bsum[t]);
        atomicAdd(&gsq[t], bsq[t]);
    }
}

// ---------------------------------------------------------------------------
// 6) BN finalize, normalize+GELU (in place), and final max-pool layer
// ---------------------------------------------------------------------------
__global__ void zero_bn_kernel(float* bnsum, float* bnsq) {
    int t = threadIdx.x;
    if (t < 128) { bnsum[t] = 0.0f; bnsq[t] = 0.0f; }
}

__global__ void bn_finalize_kernel(const float* gsum, const float* gsq,
                                   const float* g, const float* be,
                                   float* scale, float* shift, int nc, float invM) {
    int n = threadIdx.x;
    if (n < nc) {
        float mu = gsum[n] * invM;
        float var = gsq[n] * invM - mu * mu;
        float sc = g[n] * rsqrtf(var + BN_EPS);
        scale[n] = sc;
        shift[n] = be[n] - mu * sc;
    }
}

template <int NC>
__global__ __launch_bounds__(256) void norm_gelu_kernel(float* __restrict__ Y,
                                                        const float* __restrict__ scale,
                                                        const float* __restrict__ shift) {
    size_t i = (size_t)blockIdx.x * 256 + threadIdx.x;
    int n = (int)(i & (NC - 1));
    float v = Y[i] * scale[n] + shift[n];
    Y[i] = gelu_f(v);
}

__global__ __launch_bounds__(256) void final_max_kernel(const float* __restrict__ Y,
                                                        const float* __restrict__ scale,
                                                        const float* __restrict__ shift,
                                                        float* __restrict__ out) {
    int idx = blockIdx.x * 256 + threadIdx.x;      // (b, c, s) flat, c-major over s
    int b = idx >> 17;
    int rem = idx & 131071;
    int c = rem >> 10;
    int s = rem & 1023;
    float sc = scale[c], sh = shift[c];
    float best = -3.4e38f;
    size_t base = ((size_t)b * (SPTS * KSAMP) + (size_t)s * KSAMP) * 128 + c;
    #pragma unroll 4
    for (int k = 0; k < KSAMP; ++k) {
        float v = Y[base + (size_t)k * 128];
        v = gelu_f(v * sc + sh);
        best = fmaxf(best, v);
    }
    out[idx] = best;
}

// ---------------------------------------------------------------------------
extern "C" void kernel_launch(void* const* d_in, const int* in_sizes, int n_in,
                              void* d_out, int out_size, void* d_ws, size_t ws_size,
                              hipStream_t stream) {
    const float* xyz = (const float*)d_in[0];
    const float* pts = (const float*)d_in[1];
    const float* te  = (const float*)d_in[2];
    const float* wt[3] = { (const float*)d_in[3],  (const float*)d_in[9],  (const float*)d_in[15] };
    const float* bt[3] = { (const float*)d_in[4],  (const float*)d_in[10], (const float*)d_in[16] };
    const float* wc[3] = { (const float*)d_in[5],  (const float*)d_in[11], (const float*)d_in[17] };
    const float* bc[3] = { (const float*)d_in[6],  (const float*)d_in[12], (const float*)d_in[18] };
    const float* gg[3] = { (const float*)d_in[7],  (const float*)d_in[13], (const float*)d_in[19] };
    const float* be[3] = { (const float*)d_in[8],  (const float*)d_in[14], (const float*)d_in[20] };

    float* out  = (float*)d_out;                   // [0,24576): new_xyz ; then new_points
    float* ws   = (float*)d_ws;
    float* BUF0 = ws;                              // cap M*68
    float* BUF1 = ws + (size_t)MROWS * 68;         // cap M*128
    float* SM   = BUF1 + (size_t)MROWS * 128;
    float* tp0 = SM, *tp1 = SM + 544, *tp2 = SM + 1056;
    float* bias0 = SM + 1568, *bias1 = SM + 2080, *bias2 = SM + 2592;
    float* bnsum = SM + 3616, *bnsq = SM + 3744;
    float* bnscale = SM + 3872, *bnshift = SM + 4000;

    const float invM = 1.0f / (float)MROWS;

    // FPS -> new_xyz (first output segment)
    fps_kernel<<<BATCH, 256, 0, stream>>>(xyz, out);

    // time-conditioning folded into per-batch effective biases
    tp_kernel<<<5, 128, 0, stream>>>(te, wt[0], bt[0], tp0, 67);
    tp_kernel<<<4, 128, 0, stream>>>(te, wt[1], bt[1], tp1, 64);
    tp_kernel<<<4, 128, 0, stream>>>(te, wt[2], bt[2], tp2, 64);
    bias_kernel<<<4, 128, 0, stream>>>(tp0, wc[0], bc[0], bias0, 67, 64);
    bias_kernel<<<4, 128, 0, stream>>>(tp1, wc[1], bc[1], bias1, 64, 64);
    bias_kernel<<<8, 128, 0, stream>>>(tp2, wc[2], bc[2], bias2, 64, 128);

    // ball query + gather -> A0 (M x 68, col 67 zero)
    ball_gather_kernel<<<(BATCH * SPTS) / 8, 256, 0, stream>>>(xyz, pts, out, BUF0);

    const int gemmBlocks = MROWS / 64;             // 4096

    // layer 0: A0(BUF0, ld68) -> y(BUF1, M x 64); normalize in place
    zero_bn_kernel<<<1, 128, 0, stream>>>(bnsum, bnsq);
    gemm_bn_kernel<67, 68, 64><<<gemmBlocks, 128, 0, stream>>>(BUF0, wc[0], bias0, BUF1, bnsum, bnsq);
    bn_finalize_kernel<<<1, 64, 0, stream>>>(bnsum, bnsq, gg[0], be[0], bnscale, bnshift, 64, invM);
    norm_gelu_kernel<64><<<(MROWS * 64) / 256, 256, 0, stream>>>(BUF1, bnscale, bnshift);

    // layer 1: A1(BUF1, ld64) -> y(BUF0, M x 64); normalize in place
    zero_bn_kernel<<<1, 128, 0, stream>>>(bnsum, bnsq);
    gemm_bn_kernel<64, 64, 64><<<gemmBlocks, 128, 0, stream>>>(BUF1, wc[1], bias1, BUF0, bnsum, bnsq);
    bn_finalize_kernel<<<1, 64, 0, stream>>>(bnsum, bnsq, gg[1], be[1], bnscale, bnshift, 64, invM);
    norm_gelu_kernel<64><<<(MROWS * 64) / 256, 256, 0, stream>>>(BUF0, bnscale, bnshift);

    // layer 2: A2(BUF0, ld64) -> y(BUF1, M x 128); normalize+gelu+max fused
    zero_bn_kernel<<<1, 128, 0, stream>>>(bnsum, bnsq);
    gemm_bn_kernel<64, 64, 128><<<gemmBlocks, 128, 0, stream>>>(BUF0, wc[2], bias2, BUF1, bnsum, bnsq);
    bn_finalize_kernel<<<1, 128, 0, stream>>>(bnsum, bnsq, gg[2], be[2], bnscale, bnshift, 128, invM);
    final_max_kernel<<<(BATCH * 128 * SPTS) / 256, 256, 0, stream>>>(BUF1, bnscale, bnshift, out + BATCH * SPTS * 3);

    (void)in_sizes; (void)n_in; (void)out_size; (void)ws_size;
}